// VanillaRelationModule_27608049779168
// MI455X (gfx1250) — compile-verified
//
#include <hip/hip_runtime.h>
#include <hip/hip_bf16.h>

// ---------------------------------------------------------------------------
// Relation module for MI455X (gfx1250), fp32 WMMA (V_WMMA_F32_16X16X4_F32).
//
// Math restructuring vs reference:
//   rel[n,g,d] = sum_m sm[n,g,m] * (feats @ Wv_g^T)[m,d] + bv[g*64+d]
// so we precompute P = feats @ Wv^T once (2 GF) instead of the 34 GF
// einsum('ngm,mc->ngc') + grouped projection. Softmax rows sum to 1, so bv
// folds into the final GEMM bias. The geometric-attention term
// emb @ Wg^T is also computed on the WMMA pipe (per-row [1024,64]x[64,16]).
// ---------------------------------------------------------------------------

typedef float v2f __attribute__((ext_vector_type(2)));
typedef float v8f __attribute__((ext_vector_type(8)));

#define NROI 1024
#define CCH  1024
#define NGRP 16
#define GCH  64

// ---------------------------------------------------------------------------
// Per-box geometry: cx, cy, w, h, log(w), log(h)  (SoA, stride n)
// ---------------------------------------------------------------------------
__global__ void geom_kernel(const float* __restrict__ rois,
                            float* __restrict__ geom, int n) {
    int i = blockIdx.x * blockDim.x + threadIdx.x;
    if (i >= n) return;
    float x1 = rois[i * 5 + 1], y1 = rois[i * 5 + 2];
    float x2 = rois[i * 5 + 3], y2 = rois[i * 5 + 4];
    float w = x2 - x1 + 1.0f, h = y2 - y1 + 1.0f;
    geom[i]         = 0.5f * (x1 + x2);
    geom[n + i]     = 0.5f * (y1 + y2);
    geom[2 * n + i] = w;
    geom[3 * n + i] = h;
    geom[4 * n + i] = __logf(w);
    geom[5 * n + i] = __logf(h);
}

// ---------------------------------------------------------------------------
// WMMA fp32 GEMM, D = alpha * A @ B^T + bias.   A:[M,K] lda, B:[N,K] ldb.
// Block = 4 waves (128 thr), block tile 64x64, wave tile 32x32 (2x2 accs).
// Batched via blockIdx.z with element strides (used for per-group views).
// Fragment layouts per CDNA5 ISA 7.12.2:
//   A 16x4 f32: lanes 0-15 -> K={0,1}, lanes 16-31 -> K={2,3} (v2f)
//   C/D 16x16 : VGPR j -> row j (lanes 0-15) / row 8+j (lanes 16-31)
// ---------------------------------------------------------------------------
__global__ __launch_bounds__(128) void wmma_gemm_nt(
    const float* __restrict__ A, int lda, long long strideA,
    const float* __restrict__ B, int ldb, long long strideB,
    float* __restrict__ D, int ldd, long long strideD,
    const float* __restrict__ bias, long long strideBias,
    float alpha, int K) {
    A += (size_t)blockIdx.z * strideA;
    B += (size_t)blockIdx.z * strideB;
    D += (size_t)blockIdx.z * strideD;
    if (bias) bias += (size_t)blockIdx.z * strideBias;

    const int lane    = threadIdx.x & 31;
    const int wave    = threadIdx.x >> 5;
    const int tileRow = blockIdx.y * 64 + (wave >> 1) * 32;
    const int tileCol = blockIdx.x * 64 + (wave & 1) * 32;
    const int l       = lane & 15;
    const int khalf   = (lane >> 4) << 1;   // 0 or 2

    v8f acc[2][2] = {};

    const float* a0 = A + (size_t)(tileRow + l) * lda + khalf;
    const float* a1 = a0 + (size_t)16 * lda;
    const float* b0 = B + (size_t)(tileCol + l) * ldb + khalf;
    const float* b1 = b0 + (size_t)16 * ldb;

#pragma unroll 2
    for (int k = 0; k < K; k += 4) {
        v2f af0 = *(const v2f*)(a0 + k);
        v2f af1 = *(const v2f*)(a1 + k);
        v2f bf0 = *(const v2f*)(b0 + k);
        v2f bf1 = *(const v2f*)(b1 + k);
        acc[0][0] = __builtin_amdgcn_wmma_f32_16x16x4_f32(false, af0, false, bf0, (short)0, acc[0][0], false, false);
        acc[0][1] = __builtin_amdgcn_wmma_f32_16x16x4_f32(false, af0, false, bf1, (short)0, acc[0][1], false, false);
        acc[1][0] = __builtin_amdgcn_wmma_f32_16x16x4_f32(false, af1, false, bf0, (short)0, acc[1][0], false, false);
        acc[1][1] = __builtin_amdgcn_wmma_f32_16x16x4_f32(false, af1, false, bf1, (short)0, acc[1][1], false, false);
    }

    const int colOff = lane & 15;
    const int rowOff = (lane >> 4) << 3;
#pragma unroll
    for (int r = 0; r < 2; ++r)
#pragma unroll
        for (int c = 0; c < 2; ++c) {
            int col  = tileCol + c * 16 + colOff;
            float bb = bias ? bias[col] : 0.0f;
#pragma unroll
            for (int j = 0; j < 8; ++j) {
                int row = tileRow + r * 16 + rowOff + j;
                D[(size_t)row * ldd + col] = alpha * acc[r][c][j] + bb;
            }
        }
}

// ---------------------------------------------------------------------------
// WMMA fp32 GEMM, D = alpha * A @ B + bias.   A:[M,K] lda, B:[K,N] ldb.
// Same tiling; B fragment gathered row-wise (K rows striped across lanes).
// ---------------------------------------------------------------------------
__global__ __launch_bounds__(128) void wmma_gemm_nn(
    const float* __restrict__ A, int lda, long long strideA,
    const float* __restrict__ B, int ldb, long long strideB,
    float* __restrict__ D, int ldd, long long strideD,
    const float* __restrict__ bias, long long strideBias,
    float alpha, int K) {
    A += (size_t)blockIdx.z * strideA;
    B += (size_t)blockIdx.z * strideB;
    D += (size_t)blockIdx.z * strideD;
    if (bias) bias += (size_t)blockIdx.z * strideBias;

    const int lane    = threadIdx.x & 31;
    const int wave    = threadIdx.x >> 5;
    const int tileRow = blockIdx.y * 64 + (wave >> 1) * 32;
    const int tileCol = blockIdx.x * 64 + (wave & 1) * 32;
    const int l       = lane & 15;
    const int khalf   = (lane >> 4) << 1;

    v8f acc[2][2] = {};

    const float* a0 = A + (size_t)(tileRow + l) * lda + khalf;
    const float* a1 = a0 + (size_t)16 * lda;

#pragma unroll 2
    for (int k = 0; k < K; k += 4) {
        v2f af0 = *(const v2f*)(a0 + k);
        v2f af1 = *(const v2f*)(a1 + k);
        const float* bp = B + (size_t)(k + khalf) * ldb;
        v2f bf0, bf1;
        bf0.x = bp[tileCol + l];
        bf0.y = bp[(size_t)ldb + tileCol + l];
        bf1.x = bp[tileCol + 16 + l];
        bf1.y = bp[(size_t)ldb + tileCol + 16 + l];
        acc[0][0] = __builtin_amdgcn_wmma_f32_16x16x4_f32(false, af0, false, bf0, (short)0, acc[0][0], false, false);
        acc[0][1] = __builtin_amdgcn_wmma_f32_16x16x4_f32(false, af0, false, bf1, (short)0, acc[0][1], false, false);
        acc[1][0] = __builtin_amdgcn_wmma_f32_16x16x4_f32(false, af1, false, bf0, (short)0, acc[1][0], false, false);
        acc[1][1] = __builtin_amdgcn_wmma_f32_16x16x4_f32(false, af1, false, bf1, (short)0, acc[1][1], false, false);
    }

    const int colOff = lane & 15;
    const int rowOff = (lane >> 4) << 3;
#pragma unroll
    for (int r = 0; r < 2; ++r)
#pragma unroll
        for (int c = 0; c < 2; ++c) {
            int col  = tileCol + c * 16 + colOff;
            float bb = bias ? bias[col] : 0.0f;
#pragma unroll
            for (int j = 0; j < 8; ++j) {
                int row = tileRow + r * 16 + rowOff + j;
                D[(size_t)row * ldd + col] = alpha * acc[r][c][j] + bb;
            }
        }
}

// ---------------------------------------------------------------------------
// Per-row n: geometric attention weight via WMMA + softmax, in place on
// logits[g][n][m] (which holds the scaled q.k affinity on entry).
//
// Block = 512 threads = 16 waves; one block per row n. Each wave owns 4
// m-tiles of 16. A-fragments (position embedding, 16x4 f32 layout) are
// synthesized in registers: a lane's half (khalf) only ever needs pe indices
// congruent to {khalf, khalf+1} mod 4, which is exactly 16 sincos per tile,
// and the frequency 1000^{-(d+khalf)/8} factors into const(d) * kscale(khalf)
// so every fragment index is compile-time constant.
// Each lane owns group column g = lane&15; softmax over m keeps all values
// in registers with only 32-wide LDS reductions per group.
// ---------------------------------------------------------------------------
__global__ __launch_bounds__(512) void geo_softmax_wmma(
    const float* __restrict__ geom, const float* __restrict__ Wg,
    const float* __restrict__ bg, float* __restrict__ logits, int n) {
    __shared__ float red[NGRP * 33];           // padded: stride 33 avoids bank clash
    const int tid    = threadIdx.x;
    const int lane   = tid & 31;
    const int wave   = tid >> 5;               // 0..15
    const int row    = blockIdx.x;             // n index
    const int l      = lane & 15;              // group column owned by lane
    const int half   = lane >> 4;              // 0/1
    const int khalf  = half << 1;              // 0 or 2
    const int rowOff = half << 3;              // D-fragment row offset
    const int pid    = wave * 2 + half;        // 0..31 reduction participant id

    // row-n geometry (uniform across block)
    const float cxn = geom[row],  cyn = geom[n + row];
    const float inv_wn = 1.0f / geom[2 * n + row];
    const float inv_hn = 1.0f / geom[3 * n + row];
    const float lwn = geom[4 * n + row], lhn = geom[5 * n + row];

    // B fragment: B[k][g] = Wg[g][k]; lane holds col g=l, rows k=4kk+khalf{,+1}
    v2f bfrag[16];
#pragma unroll
    for (int kk = 0; kk < 16; ++kk) {
        bfrag[kk].x = Wg[l * 64 + kk * 4 + khalf];
        bfrag[kk].y = Wg[l * 64 + kk * 4 + khalf + 1];
    }
    const float bgv = bg[l];

    // 1/dim_mat split: 1000^{-(d+khalf)/8} = invd[d] * kscale(khalf)
    const float invd[4] = {1.0f,                    // d=0
                           0.42169650342906526f,    // d=1 : 1000^-1/8
                           0.03162277660168379f,    // d=4 : 1000^-1/2
                           0.013335214321633238f};  // d=5 : 1000^-5/8
    const float kscale = half ? 0.17782794100389228f /*1000^-1/4*/ : 1.0f;

    float lgv[32];   // 4 tiles x 8 rows of owned logits (col g=l)

#pragma unroll
    for (int t = 0; t < 4; ++t) {
        const int tile_m = (t * 16 + wave) * 16;   // tiles 0..63 over m
        const int m = tile_m + l;                  // lane-pair shares box m
        float pm[4];
        pm[0] = __logf(fmaxf(fabsf((cxn - geom[m]) * inv_wn), 1e-3f));
        pm[1] = __logf(fmaxf(fabsf((cyn - geom[n + m]) * inv_hn), 1e-3f));
        pm[2] = lwn - geom[4 * n + m];
        pm[3] = lhn - geom[5 * n + m];

        // A fragment: pe = f*16 + j (sin) / f*16+8+j (cos), j = d + khalf,
        // d in {0,1,4,5}; kk = pe>>2 resolves to 4f + (d>=4) (+2 for cos).
        v2f afrag[16];
#pragma unroll
        for (int f = 0; f < 4; ++f)
#pragma unroll
            for (int a = 0; a < 4; ++a) {
                float s, c;
                __sincosf(100.0f * pm[f] * invd[a] * kscale, &s, &c);
                const int kks = 4 * f + (a >> 1);  // sin slot; cos slot +2
                if (a & 1) { afrag[kks].y = s; afrag[kks + 2].y = c; }
                else       { afrag[kks].x = s; afrag[kks + 2].x = c; }
            }

        v8f acc = {};
#pragma unroll
        for (int kk = 0; kk < 16; ++kk)
            acc = __builtin_amdgcn_wmma_f32_16x16x4_f32(
                false, afrag[kk], false, bfrag[kk], (short)0, acc, false, false);

        // D row m' = tile_m + rowOff + j, col g = l:
        // log(clip(relu(aw),1e-6)) + aff  ==  log(max(aw,1e-6)) + aff
#pragma unroll
        for (int j = 0; j < 8; ++j) {
            const int mm = tile_m + rowOff + j;
            float aw = acc[j] + bgv;
            lgv[t * 8 + j] = __logf(fmaxf(aw, 1e-6f)) +
                             logits[((size_t)l * n + row) * n + mm];
        }
    }

    // ---- softmax over m (1024 values per g, spread over 32 lanes x 32) ----
    float mx = lgv[0];
#pragma unroll
    for (int i = 1; i < 32; ++i) mx = fmaxf(mx, lgv[i]);
    red[l * 33 + pid] = mx;
    __syncthreads();
#pragma unroll
    for (int i = 0; i < 32; ++i) mx = fmaxf(mx, red[l * 33 + i]);

    float lsum = 0.0f;
#pragma unroll
    for (int i = 0; i < 32; ++i) {
        lgv[i] = __expf(lgv[i] - mx);
        lsum += lgv[i];
    }
    __syncthreads();
    red[l * 33 + pid] = lsum;
    __syncthreads();
    float tot = 0.0f;
#pragma unroll
    for (int i = 0; i < 32; ++i) tot += red[l * 33 + i];
    const float inv = 1.0f / tot;

#pragma unroll
    for (int t = 0; t < 4; ++t) {
        const int base = (t * 16 + wave) * 16 + rowOff;
#pragma unroll
        for (int j = 0; j < 8; ++j)
            logits[((size_t)l * n + row) * n + base + j] = lgv[t * 8 + j] * inv;
    }
}

// ---------------------------------------------------------------------------
extern "C" void kernel_launch(void* const* d_in, const int* in_sizes, int n_in,
                              void* d_out, int out_size, void* d_ws, size_t ws_size,
                              hipStream_t stream) {
    (void)in_sizes; (void)n_in; (void)out_size; (void)ws_size;
    const float* feats = (const float*)d_in[0];
    const float* rois  = (const float*)d_in[1];
    const float* Wq    = (const float*)d_in[2];
    const float* bq    = (const float*)d_in[3];
    const float* Wk    = (const float*)d_in[4];
    const float* bk    = (const float*)d_in[5];
    const float* Wg    = (const float*)d_in[6];
    const float* bg    = (const float*)d_in[7];
    const float* Wv    = (const float*)d_in[8];
    const float* bv    = (const float*)d_in[9];
    float* out = (float*)d_out;

    const int n = NROI, c = CCH, g = NGRP, gc = GCH;

    // workspace layout (floats): geom | q | k | P | logits[g][n][m]
    float* ws     = (float*)d_ws;
    float* geom   = ws;                       // 6*n (padded to 8192)
    float* q      = ws + 8192;                // n*c
    float* kbuf   = q + (size_t)n * c;        // n*c
    float* P      = kbuf + (size_t)n * c;     // n*c
    float* logits = P + (size_t)n * c;        // g*n*n (64 MB)

    geom_kernel<<<dim3((n + 255) / 256), dim3(256), 0, stream>>>(rois, geom, n);

    const dim3 blk(128);
    // q = feats @ Wq^T + bq ; k = feats @ Wk^T + bk ; P = feats @ Wv^T
    wmma_gemm_nt<<<dim3(16, 16, 1), blk, 0, stream>>>(
        feats, c, 0, Wq, c, 0, q, c, 0, bq, 0, 1.0f, c);
    wmma_gemm_nt<<<dim3(16, 16, 1), blk, 0, stream>>>(
        feats, c, 0, Wk, c, 0, kbuf, c, 0, bk, 0, 1.0f, c);
    wmma_gemm_nt<<<dim3(16, 16, 1), blk, 0, stream>>>(
        feats, c, 0, Wv, c, 0, P, c, 0, nullptr, 0, 1.0f, c);

    // aff[g][n][m] = (q_g @ k_g^T) / sqrt(gc) ; per-group strided views
    wmma_gemm_nt<<<dim3(16, 16, g), blk, 0, stream>>>(
        q, c, gc, kbuf, c, gc, logits, n, (long long)n * n,
        nullptr, 0, 0.125f, gc);

    // geometric term via WMMA + softmax over m (in place)
    geo_softmax_wmma<<<dim3(n), dim3(512), 0, stream>>>(geom, Wg, bg, logits, n);

    // rel_g = sm_g @ P_g + bv_g   (softmax rows sum to 1 -> bv folds in)
    wmma_gemm_nn<<<dim3(1, 16, g), blk, 0, stream>>>(
        logits, n, (long long)n * n, P, c, gc, out, c, gc, bv, gc, 1.0f, n);
}